// GraphNeuralNetwork_38955353375502
// MI455X (gfx1250) — compile-verified
//
#include <hip/hip_runtime.h>
#include <math.h>

// ---------------- problem constants (match reference) ----------------
#define NN     20000
#define F_IN   256
#define HID    256
#define EMB    128
#define HEADS  8
#define NEG_SLOPE 0.2f

// ---------------- WMMA types ----------------
typedef __attribute__((ext_vector_type(2))) float v2f;
typedef __attribute__((ext_vector_type(8))) float v8f;

// ---------------- helpers ----------------
__device__ __forceinline__ float atomicMaxF(float* addr, float val) {
  if (val >= 0.f)
    return __int_as_float(atomicMax((int*)addr, __float_as_int(val)));
  else
    return __uint_as_float(atomicMin((unsigned int*)addr, __float_as_uint(val)));
}

// CDNA5 async global->LDS copy (ASYNCcnt-tracked), 16B per active lane.
__device__ __forceinline__ void async_copy_b128(const void* gptr, unsigned ldsaddr) {
  asm volatile("global_load_async_to_lds_b128 %0, %1, off"
               :: "v"(ldsaddr), "v"((unsigned long long)gptr)
               : "memory");
}
__device__ __forceinline__ void wait_async0() {
  asm volatile("s_wait_asynccnt 0x0" ::: "memory");
}

// =====================================================================
// Fused GEMM:  C[N x M] = A[N x K] @ W[K x M] (+bias) (relu?)
// fp32 WMMA 16x16x4.  Block = 320 threads = 10 waves, tile 160x16.
// Double-buffered async global->LDS staging overlapped with WMMA.
// =====================================================================
#define TM  160
#define BK  32
#define BKP 34   // padded row stride (even => 8B-aligned b64 fragment loads)

__global__ __launch_bounds__(320) void gemm_wmma_f32(
    const float* __restrict__ A, const float* __restrict__ W,
    const float* __restrict__ bias, float* __restrict__ C,
    int N, int K, int M, int hasBias, int doRelu)
{
  __shared__ __align__(16) float As[2][TM][BKP];  // row-major: As[b][m][k]
  __shared__ __align__(16) float Ws[2][BK][16];   // k-major:   Ws[b][k][n]

  const int tid  = threadIdx.x;
  const int wv   = tid >> 5;       // 0..9
  const int lane = tid & 31;
  const int l15  = lane & 15;
  const int hsel = lane >> 4;      // 0: K pair {0,1}, 1: K pair {2,3}
  const int row0 = blockIdx.x * TM;
  const int n0   = blockIdx.y * 16;

  // staging assignment (constant per thread)
  const int ar   = tid >> 1;       // A row 0..159
  const int ah   = tid & 1;        // which 16-col half of the 32-col chunk
  const int wr   = tid >> 2;       // W row 0..31 (tid < 128)
  const int wcg  = tid & 3;        // W col group

  const float* aRowG = A + (size_t)(row0 + ar) * K + ah * 16;

  v8f acc = {};

  // issue async copies of chunk starting at kc into buffer `buf`
  auto issue = [&](int kc, int buf) {
    const float* ap = aRowG + kc;
    const unsigned albase = (unsigned)(size_t)&As[buf][ar][ah * 16];
#pragma unroll
    for (int i = 0; i < 4; ++i)
      async_copy_b128(ap + i * 4, albase + i * 16);
    if (tid < 128) {
      async_copy_b128(W + (size_t)(kc + wr) * M + n0 + wcg * 4,
                      (unsigned)(size_t)&Ws[buf][wr][wcg * 4]);
    }
  };

  int cur = 0;
  issue(0, 0);
  wait_async0();
  __syncthreads();

  const int m = wv * 16 + l15;
  for (int kc = 0;;) {
    const int nxt = cur ^ 1;
    const int kn  = kc + BK;
    if (kn < K) issue(kn, nxt);        // overlap next-tile copy with compute

    const float* arow = &As[cur][m][0];
#pragma unroll
    for (int kk = 0; kk < BK; kk += 4) {
      const int k0 = kk + hsel * 2;
      v2f a = *(const v2f*)(arow + k0);
      v2f b;
      b.x = Ws[cur][k0][l15];
      b.y = Ws[cur][k0 + 1][l15];
      acc = __builtin_amdgcn_wmma_f32_16x16x4_f32(
          false, a, false, b, (short)0, acc, false, false);
    }

    kc = kn;
    if (kc >= K) break;
    wait_async0();        // own async writes to `nxt` done...
    __syncthreads();      // ...and visible to all waves; reads of `cur` done
    cur = nxt;
  }

  const float bv = hasBias ? bias[n0 + l15] : 0.f;
  const int rbase = row0 + wv * 16 + (hsel ? 8 : 0);
#pragma unroll
  for (int v = 0; v < 8; ++v) {
    float x = acc[v] + bv;
    if (doRelu) x = fmaxf(x, 0.f);
    C[(size_t)(rbase + v) * M + n0 + l15] = x;
  }
}

// =====================================================================
// small utility kernels
// =====================================================================
__global__ void fill_kernel(float* __restrict__ p, float v, long long n) {
  long long stride = (long long)gridDim.x * blockDim.x;
  for (long long t = (long long)blockIdx.x * blockDim.x + threadIdx.x; t < n; t += stride)
    p[t] = v;
}

__global__ void deg_kernel(const long long* __restrict__ ei, float* __restrict__ deg, int E) {
  long long stride = (long long)gridDim.x * blockDim.x;
  for (long long t = (long long)blockIdx.x * blockDim.x + threadIdx.x; t < E; t += stride)
    unsafeAtomicAdd(&deg[(int)ei[E + t]], 1.0f);
}

__global__ void dinv_kernel(float* __restrict__ deg, int n) {
  long long stride = (long long)gridDim.x * blockDim.x;
  for (long long t = (long long)blockIdx.x * blockDim.x + threadIdx.x; t < n; t += stride)
    deg[t] = rsqrtf(fmaxf(deg[t], 1.0f));
}

// GCN edge scatter: out[dst] += P[src] * dinv[src]*dinv[dst]   (float4 lanes)
__global__ void gcn_scatter(const float* __restrict__ P, float* __restrict__ out,
                            const long long* __restrict__ ei,
                            const float* __restrict__ dinv, int E, int C) {
  const int c4 = C >> 2;
  const long long total = (long long)E * c4;
  long long stride = (long long)gridDim.x * blockDim.x;
  for (long long t = (long long)blockIdx.x * blockDim.x + threadIdx.x; t < total; t += stride) {
    const int e  = (int)(t / c4);
    const int f4 = (int)(t % c4);
    const int s = (int)ei[e];
    const int d = (int)ei[E + e];
    const float c = dinv[s] * dinv[d];
    const float4 v = *(const float4*)(P + (size_t)s * C + f4 * 4);
    float* o = out + (size_t)d * C + f4 * 4;
    unsafeAtomicAdd(o + 0, v.x * c);
    unsafeAtomicAdd(o + 1, v.y * c);
    unsafeAtomicAdd(o + 2, v.z * c);
    unsafeAtomicAdd(o + 3, v.w * c);
  }
}

// out = relu(out + P*dinv^2 (self loop) + bias)
__global__ void gcn_finalize(float* __restrict__ out, const float* __restrict__ P,
                             const float* __restrict__ dinv,
                             const float* __restrict__ bias, int N, int C) {
  const long long total = (long long)N * C;
  long long stride = (long long)gridDim.x * blockDim.x;
  for (long long t = (long long)blockIdx.x * blockDim.x + threadIdx.x; t < total; t += stride) {
    const int n = (int)(t / C);
    const int f = (int)(t % C);
    const float di = dinv[n];
    out[t] = fmaxf(out[t] + P[t] * di * di + bias[f], 0.f);
  }
}

// per-node, per-head attention coefficients; also init m=-inf, denom=0
__global__ __launch_bounds__(256) void att_coeff(
    const float* __restrict__ HG, const float* __restrict__ att_src,
    const float* __restrict__ att_dst, float* __restrict__ a_s,
    float* __restrict__ a_d, float* __restrict__ m, float* __restrict__ denom, int N) {
  const int n = blockIdx.x;
  const int h = threadIdx.x >> 5;     // wave == head
  const int lane = threadIdx.x & 31;
  const float* hp = HG + (size_t)n * (HEADS * EMB) + h * EMB + lane * 4;
  const float* as = att_src + h * EMB + lane * 4;
  const float* ad = att_dst + h * EMB + lane * 4;
  float ss = 0.f, sd = 0.f;
#pragma unroll
  for (int j = 0; j < 4; ++j) { float v = hp[j]; ss += v * as[j]; sd += v * ad[j]; }
  for (int o = 16; o > 0; o >>= 1) { ss += __shfl_down(ss, o, 32); sd += __shfl_down(sd, o, 32); }
  if (lane == 0) {
    a_s[n * HEADS + h] = ss;
    a_d[n * HEADS + h] = sd;
    m[n * HEADS + h] = -INFINITY;
    denom[n * HEADS + h] = 0.f;
  }
}

// pass 1: e = leaky_relu(a_s[src]+a_d[dst]); segment max
__global__ void gat_edge_max(const float* __restrict__ a_s, const float* __restrict__ a_d,
                             const long long* __restrict__ ei, float* __restrict__ m,
                             float* __restrict__ ebuf, int E, int N) {
  const long long total = (long long)(E + N) * HEADS;
  long long stride = (long long)gridDim.x * blockDim.x;
  for (long long t = (long long)blockIdx.x * blockDim.x + threadIdx.x; t < total; t += stride) {
    const int e = (int)(t >> 3);
    const int h = (int)(t & 7);
    int s, d;
    if (e < E) { s = (int)ei[e]; d = (int)ei[E + e]; } else { s = d = e - E; }
    float x = a_s[s * HEADS + h] + a_d[d * HEADS + h];
    x = (x > 0.f) ? x : NEG_SLOPE * x;
    ebuf[t] = x;
    atomicMaxF(&m[d * HEADS + h], x);
  }
}

// pass 2: ex = exp(e - m[dst]); segment sum
__global__ void gat_edge_sum(const long long* __restrict__ ei, const float* __restrict__ m,
                             float* __restrict__ denom, float* __restrict__ ebuf, int E, int N) {
  const long long total = (long long)(E + N) * HEADS;
  long long stride = (long long)gridDim.x * blockDim.x;
  for (long long t = (long long)blockIdx.x * blockDim.x + threadIdx.x; t < total; t += stride) {
    const int e = (int)(t >> 3);
    const int h = (int)(t & 7);
    const int d = (e < E) ? (int)ei[E + e] : (e - E);
    const float ex = expf(ebuf[t] - m[d * HEADS + h]);
    ebuf[t] = ex;
    unsafeAtomicAdd(&denom[d * HEADS + h], ex);
  }
}

// pass 3: out[dst] += HG[src] * alpha  (per head, float4 lanes)
__global__ void gat_scatter(const float* __restrict__ HG, float* __restrict__ out,
                            const long long* __restrict__ ei, const float* __restrict__ ebuf,
                            const float* __restrict__ denom, int E, int N) {
  const int c4 = EMB >> 2;                       // 32
  const long long total = (long long)(E + N) * HEADS * c4;
  long long stride = (long long)gridDim.x * blockDim.x;
  for (long long t = (long long)blockIdx.x * blockDim.x + threadIdx.x; t < total; t += stride) {
    const int e   = (int)(t / (HEADS * c4));
    const int rem = (int)(t % (HEADS * c4));
    const int h   = rem >> 5;
    const int f4  = rem & 31;
    int s, d;
    if (e < E) { s = (int)ei[e]; d = (int)ei[E + e]; } else { s = d = e - E; }
    const float alpha = ebuf[(size_t)e * HEADS + h] / denom[d * HEADS + h];
    const float4 v = *(const float4*)(HG + (size_t)s * (HEADS * EMB) + h * EMB + f4 * 4);
    float* o = out + (size_t)d * (HEADS * EMB) + h * EMB + f4 * 4;
    unsafeAtomicAdd(o + 0, v.x * alpha);
    unsafeAtomicAdd(o + 1, v.y * alpha);
    unsafeAtomicAdd(o + 2, v.z * alpha);
    unsafeAtomicAdd(o + 3, v.w * alpha);
  }
}

__global__ void add_rowvec(float* __restrict__ out, const float* __restrict__ b,
                           int N, int C) {
  const long long total = (long long)N * C;
  long long stride = (long long)gridDim.x * blockDim.x;
  for (long long t = (long long)blockIdx.x * blockDim.x + threadIdx.x; t < total; t += stride)
    out[t] += b[(int)(t % C)];
}

// logits + sigmoid: one wave per node (z[n] . Wh2 + bh2)
__global__ __launch_bounds__(256) void head_kernel(
    const float* __restrict__ z, const float* __restrict__ Wh2,
    const float* __restrict__ bh2, float* __restrict__ wbuf, int N) {
  const int wv = threadIdx.x >> 5;
  const int lane = threadIdx.x & 31;
  const int n = blockIdx.x * 8 + wv;
  if (n >= N) return;
  float s = 0.f;
  const float* zp = z + (size_t)n * HID + lane * 8;
  const float* wp = Wh2 + lane * 8;
#pragma unroll
  for (int j = 0; j < 8; ++j) s += zp[j] * wp[j];
  for (int o = 16; o > 0; o >>= 1) s += __shfl_down(s, o, 32);
  if (lane == 0) {
    const float lg = s + bh2[0];
    wbuf[n] = 1.f / (1.f + expf(-lg));
  }
}

__global__ void init_scal(float* scal) { scal[0] = -INFINITY; scal[1] = 0.f; }

__global__ __launch_bounds__(256) void reduce_max_k(const float* __restrict__ w,
                                                    float* __restrict__ scal, int N) {
  __shared__ float sm[256];
  float v = -INFINITY;
  long long stride = (long long)gridDim.x * blockDim.x;
  for (long long t = (long long)blockIdx.x * blockDim.x + threadIdx.x; t < N; t += stride)
    v = fmaxf(v, w[t]);
  sm[threadIdx.x] = v; __syncthreads();
  for (int s = 128; s > 0; s >>= 1) {
    if (threadIdx.x < s) sm[threadIdx.x] = fmaxf(sm[threadIdx.x], sm[threadIdx.x + s]);
    __syncthreads();
  }
  if (threadIdx.x == 0) atomicMaxF(&scal[0], sm[0]);
}

__global__ __launch_bounds__(256) void exp_sum_k(const float* __restrict__ w,
                                                 const float* __restrict__ scal,
                                                 float* __restrict__ outp,
                                                 float* __restrict__ scalsum, int N) {
  __shared__ float sm[256];
  const float mx = scal[0];
  float acc = 0.f;
  long long stride = (long long)gridDim.x * blockDim.x;
  for (long long t = (long long)blockIdx.x * blockDim.x + threadIdx.x; t < N; t += stride) {
    const float e = expf(w[t] - mx);
    outp[t] = e;
    acc += e;
  }
  sm[threadIdx.x] = acc; __syncthreads();
  for (int s = 128; s > 0; s >>= 1) {
    if (threadIdx.x < s) sm[threadIdx.x] += sm[threadIdx.x + s];
    __syncthreads();
  }
  if (threadIdx.x == 0) unsafeAtomicAdd(&scalsum[1], sm[0]);
}

__global__ void normalize_k(float* __restrict__ outp, const float* __restrict__ scal, int N) {
  const float inv = 1.f / scal[1];
  long long stride = (long long)gridDim.x * blockDim.x;
  for (long long t = (long long)blockIdx.x * blockDim.x + threadIdx.x; t < N; t += stride)
    outp[t] *= inv;
}

// =====================================================================
// host launcher
// =====================================================================
extern "C" void kernel_launch(void* const* d_in, const int* in_sizes, int n_in,
                              void* d_out, int out_size, void* d_ws, size_t ws_size,
                              hipStream_t stream) {
  const float*      x    = (const float*)d_in[0];
  const long long*  ei   = (const long long*)d_in[1];
  const float*      W1   = (const float*)d_in[2];
  const float*      b1   = (const float*)d_in[3];
  const float*      W2   = (const float*)d_in[4];
  const float*      b2   = (const float*)d_in[5];
  const float*      W3   = (const float*)d_in[6];
  const float*      b3   = (const float*)d_in[7];
  const float*      Wg   = (const float*)d_in[8];
  const float*      atS  = (const float*)d_in[9];
  const float*      atD  = (const float*)d_in[10];
  const float*      bg   = (const float*)d_in[11];
  const float*      Wh1  = (const float*)d_in[12];
  const float*      bh1  = (const float*)d_in[13];
  const float*      Wh2  = (const float*)d_in[14];
  const float*      bh2  = (const float*)d_in[15];
  float*            out  = (float*)d_out;

  const int N = NN;
  const int E = in_sizes[1] / 2;

  // ---- workspace layout (floats) ----
  float* ws = (float*)d_ws;
  size_t o = 0;
  float* G0   = ws + o; o += (size_t)N * 1024;   // GEMM output scratch (max width)
  float* G1   = ws + o; o += (size_t)N * HID;    // H1, later z
  float* G2   = ws + o; o += (size_t)N * HID;    // H2
  float* G3   = ws + o; o += (size_t)N * EMB;    // H3
  float* G4   = ws + o; o += (size_t)N * 1024;   // HG
  float* G5   = ws + o; o += (size_t)N * 1024;   // GAT output
  float* dinv = ws + o; o += (size_t)N;          // degree -> dinv (in place)
  float* a_s  = ws + o; o += (size_t)N * HEADS;
  float* a_d  = ws + o; o += (size_t)N * HEADS;
  float* mbuf = ws + o; o += (size_t)N * HEADS;
  float* dbuf = ws + o; o += (size_t)N * HEADS;
  float* ebuf = ws + o; o += (size_t)(E + N) * HEADS;
  float* wbuf = ws + o; o += (size_t)N;
  float* scal = ws + o; o += 2;

  const int GSB = 4096;   // grid for grid-stride kernels
  const int TPB = 256;
  dim3 gemmBlk(320);

  // 1) degrees -> dinv
  fill_kernel<<<GSB, TPB, 0, stream>>>(dinv, 1.0f, (long long)N);
  deg_kernel<<<GSB, TPB, 0, stream>>>(ei, dinv, E);
  dinv_kernel<<<GSB, TPB, 0, stream>>>(dinv, N);

  // 2) GCN layer 1: P = x@W1 ; H1 = relu(scatter + self + b1)
  gemm_wmma_f32<<<dim3(N / TM, HID / 16), gemmBlk, 0, stream>>>(x, W1, b1, G0, N, F_IN, HID, 0, 0);
  fill_kernel<<<GSB, TPB, 0, stream>>>(G1, 0.f, (long long)N * HID);
  gcn_scatter<<<GSB, TPB, 0, stream>>>(G0, G1, ei, dinv, E, HID);
  gcn_finalize<<<GSB, TPB, 0, stream>>>(G1, G0, dinv, b1, N, HID);

  // 3) GCN layer 2
  gemm_wmma_f32<<<dim3(N / TM, HID / 16), gemmBlk, 0, stream>>>(G1, W2, b2, G0, N, HID, HID, 0, 0);
  fill_kernel<<<GSB, TPB, 0, stream>>>(G2, 0.f, (long long)N * HID);
  gcn_scatter<<<GSB, TPB, 0, stream>>>(G0, G2, ei, dinv, E, HID);
  gcn_finalize<<<GSB, TPB, 0, stream>>>(G2, G0, dinv, b2, N, HID);

  // 4) GCN layer 3 (HID -> EMB)
  gemm_wmma_f32<<<dim3(N / TM, EMB / 16), gemmBlk, 0, stream>>>(G2, W3, b3, G0, N, HID, EMB, 0, 0);
  fill_kernel<<<GSB, TPB, 0, stream>>>(G3, 0.f, (long long)N * EMB);
  gcn_scatter<<<GSB, TPB, 0, stream>>>(G0, G3, ei, dinv, E, EMB);
  gcn_finalize<<<GSB, TPB, 0, stream>>>(G3, G0, dinv, b3, N, EMB);

  // 5) GAT: HG = H3 @ Wg  (EMB -> HEADS*EMB = 1024)
  gemm_wmma_f32<<<dim3(N / TM, (HEADS * EMB) / 16), gemmBlk, 0, stream>>>(
      G3, Wg, bg, G4, N, EMB, HEADS * EMB, 0, 0);
  att_coeff<<<N, 256, 0, stream>>>(G4, atS, atD, a_s, a_d, mbuf, dbuf, N);
  gat_edge_max<<<GSB, TPB, 0, stream>>>(a_s, a_d, ei, mbuf, ebuf, E, N);
  gat_edge_sum<<<GSB, TPB, 0, stream>>>(ei, mbuf, dbuf, ebuf, E, N);
  fill_kernel<<<GSB, TPB, 0, stream>>>(G5, 0.f, (long long)N * 1024);
  gat_scatter<<<GSB, TPB, 0, stream>>>(G4, G5, ei, ebuf, dbuf, E, N);
  add_rowvec<<<GSB, TPB, 0, stream>>>(G5, bg, N, HEADS * EMB);

  // 6) MLP head: z = relu(G5 @ Wh1 + bh1)  (1024 -> 256), reuse G1
  gemm_wmma_f32<<<dim3(N / TM, HID / 16), gemmBlk, 0, stream>>>(
      G5, Wh1, bh1, G1, N, HEADS * EMB, HID, 1, 1);

  // 7) logits -> sigmoid -> softmax over N
  head_kernel<<<(N + 7) / 8, 256, 0, stream>>>(G1, Wh2, bh2, wbuf, N);
  init_scal<<<1, 1, 0, stream>>>(scal);
  reduce_max_k<<<256, 256, 0, stream>>>(wbuf, scal, N);
  exp_sum_k<<<256, 256, 0, stream>>>(wbuf, scal, out, scal, N);
  normalize_k<<<GSB, TPB, 0, stream>>>(out, scal, N);
}